// LigerFusedLinearJSD_70325794505339
// MI455X (gfx1250) — compile-verified
//
#include <hip/hip_runtime.h>
#include <hip/hip_bf16.h>
#include <cstdint>

// Problem constants (match reference)
#define BT   2048
#define H_S  2048
#define H_T  4096
#define VCAB 32000
#define IGNORE_INDEX (-100LL)
#define JSD_BETA 0.5f

typedef __attribute__((ext_vector_type(2))) float v2f;
typedef __attribute__((ext_vector_type(8))) float v8f;

// ---------------------------------------------------------------------------
// GEMM: out[M=BT, N=VCAB] = X[M,H] * W[N,H]^T  (both row-major, K=H contiguous)
// Block: 128 threads (4 waves). Tile 64(M) x 64(N), BK=32.
// Wave w computes M rows [w*16, w*16+16) x all four 16-wide N subtiles.
// f32 end-to-end (memory-bound problem) via V_WMMA_F32_16X16X4_F32.
// Tile staging: GLOBAL_LOAD_ASYNC_TO_LDS_B128 (ASYNCcnt) with LDS double
// buffering so the next K-tile streams in while WMMAs consume the current one.
// Grid: blockIdx.x = M tile (fast), blockIdx.y = N tile (slow) so that all M
// blocks sharing a weight tile dispatch consecutively -> weight tile is read
// from HBM once and served from L2; activations stay L2-resident.
// ---------------------------------------------------------------------------
template<int H>
__global__ __launch_bounds__(128) void gemm_logits_f32(
    const float* __restrict__ X, const float* __restrict__ W,
    float* __restrict__ out)
{
    constexpr int BK = 32;
    // stride 36 floats: 144B rows (16B aligned for b128, bank-skewed)
    __shared__ float As[2][64][36];
    __shared__ float Bs[2][64][36];

    const int tid  = threadIdx.x;
    const int lane = tid & 31;
    const int wave = tid >> 5;      // 0..3
    const int half = lane >> 4;     // 0 or 1 (lane group)
    const int l16  = lane & 15;
    const int m0   = blockIdx.x * 64;   // M tile origin (fast grid dim)
    const int n0   = blockIdx.y * 64;   // N (vocab) tile origin (slow grid dim)
    const int wm   = wave * 16;         // wave's M offset inside tile

    // Issue one K-tile of async global->LDS copies (8 x b128 per thread:
    // 64 rows x 32 floats for A and for B, 128 threads cooperative).
    auto issue_tile = [&](int buf, int k0) {
        #pragma unroll
        for (int i = 0; i < 4; ++i) {
            int idx = tid + i * 128;
            int r   = idx >> 3;          // 0..63
            int c4  = (idx & 7) * 4;     // 0,4,...,28
            unsigned long long ga = (unsigned long long)&X[(size_t)(m0 + r) * H + k0 + c4];
            unsigned long long gb = (unsigned long long)&W[(size_t)(n0 + r) * H + k0 + c4];
            unsigned la = (unsigned)(size_t)&As[buf][r][c4];   // LDS byte offset (addr[31:0])
            unsigned lb = (unsigned)(size_t)&Bs[buf][r][c4];
            asm volatile("global_load_async_to_lds_b128 %0, %1, off"
                         :: "v"(la), "v"(ga) : "memory");
            asm volatile("global_load_async_to_lds_b128 %0, %1, off"
                         :: "v"(lb), "v"(gb) : "memory");
        }
    };

    const v8f vzero = {0.f,0.f,0.f,0.f,0.f,0.f,0.f,0.f};
    v8f acc[4];
    #pragma unroll
    for (int s = 0; s < 4; ++s) acc[s] = vzero;

    // Prologue: fill buffer 0.
    issue_tile(0, 0);
    asm volatile("s_wait_asynccnt 0x0" ::: "memory");
    __syncthreads();

    int cur = 0;
    for (int k0 = 0; k0 < H; k0 += BK) {
        const int nxt = cur ^ 1;
        // Overlap: start streaming the next K-tile while we compute.
        if (k0 + BK < H) {
            issue_tile(nxt, k0 + BK);
            // Prefetch the tile after that into L2 (global_prefetch_b8).
            if (k0 + 2 * BK < H) {
                __builtin_prefetch(&X[(size_t)(m0 + (tid >> 1)) * H + k0 + 2 * BK], 0, 3);
                __builtin_prefetch(&W[(size_t)(n0 + (tid >> 1)) * H + k0 + 2 * BK], 0, 3);
            }
        }

        // Inner product: 8 K-steps of 4, 4 N-subtiles -> 32 WMMAs per wave.
        #pragma unroll
        for (int kk = 0; kk < BK; kk += 4) {
            // A 16x4 f32 fragment: lanes 0-15 -> K=kk+0/1 ; lanes 16-31 -> K=kk+2/3
            v2f a;
            a.x = As[cur][wm + l16][kk + 2 * half + 0];
            a.y = As[cur][wm + l16][kk + 2 * half + 1];
            #pragma unroll
            for (int s = 0; s < 4; ++s) {
                // B 4x16 fragment (B = W^T): VGPR v, group g -> K = kk + v + 2g, N = lane&15
                v2f b;
                b.x = Bs[cur][s * 16 + l16][kk + 2 * half + 0];
                b.y = Bs[cur][s * 16 + l16][kk + 2 * half + 1];
                acc[s] = __builtin_amdgcn_wmma_f32_16x16x4_f32(
                    false, a, false, b, (short)0, acc[s], false, false);
            }
        }

        // Next buffer's async copies must have landed, and every wave must be
        // done reading `cur` before it is overwritten next iteration.
        asm volatile("s_wait_asynccnt 0x0" ::: "memory");
        __syncthreads();
        cur = nxt;
    }

    // D layout: VGPR r -> M = r + 8*half ; N = lane&15
    #pragma unroll
    for (int s = 0; s < 4; ++s) {
        #pragma unroll
        for (int r = 0; r < 8; ++r) {
            out[(size_t)(m0 + wm + r + 8 * half) * VCAB + n0 + s * 16 + l16] = acc[s][r];
        }
    }
}

// ---------------------------------------------------------------------------
// Per-row JSD: one 256-thread block per row. 3 strided passes over V=32000
// (max, sum-exp, JSD element sum); rows are L2-resident between passes.
// ---------------------------------------------------------------------------
__global__ __launch_bounds__(256) void jsd_rows(
    const float* __restrict__ s_logits, const float* __restrict__ t_logits,
    const long long* __restrict__ labels, float* __restrict__ accum)
{
    const int row = blockIdx.x;
    const int tid = threadIdx.x;
    const float* __restrict__ srow = s_logits + (size_t)row * VCAB;
    const float* __restrict__ trow = t_logits + (size_t)row * VCAB;

    __shared__ float redA[256];
    __shared__ float redB[256];

    // Pass 1: row maxima
    float smax = -3.402823466e38f, tmax = -3.402823466e38f;
    for (int i = tid; i < VCAB; i += 256) {
        smax = fmaxf(smax, srow[i]);
        tmax = fmaxf(tmax, trow[i]);
    }
    redA[tid] = smax; redB[tid] = tmax;
    __syncthreads();
    for (int s = 128; s > 0; s >>= 1) {
        if (tid < s) {
            redA[tid] = fmaxf(redA[tid], redA[tid + s]);
            redB[tid] = fmaxf(redB[tid], redB[tid + s]);
        }
        __syncthreads();
    }
    smax = redA[0]; tmax = redB[0];
    __syncthreads();

    // Pass 2: partition sums
    float ssum = 0.f, tsum = 0.f;
    for (int i = tid; i < VCAB; i += 256) {
        ssum += __expf(srow[i] - smax);
        tsum += __expf(trow[i] - tmax);
    }
    redA[tid] = ssum; redB[tid] = tsum;
    __syncthreads();
    for (int s = 128; s > 0; s >>= 1) {
        if (tid < s) { redA[tid] += redA[tid + s]; redB[tid] += redB[tid + s]; }
        __syncthreads();
    }
    const float slse = smax + __logf(redA[0]);   // log Z_student
    const float tlse = tmax + __logf(redB[0]);   // log Z_teacher
    __syncthreads();

    // Pass 3: generalized JSD elements (beta = 0.5)
    float jsd = 0.f;
    for (int i = tid; i < VCAB; i += 256) {
        float slp = srow[i] - slse;   // log Q
        float tlp = trow[i] - tlse;   // log P
        float Q = __expf(slp);
        float P = __expf(tlp);
        float logM = __logf(JSD_BETA * P + (1.0f - JSD_BETA) * Q);
        jsd += JSD_BETA * P * (tlp - logM) + (1.0f - JSD_BETA) * Q * (slp - logM);
    }
    redA[tid] = jsd;
    __syncthreads();
    for (int s = 128; s > 0; s >>= 1) {
        if (tid < s) redA[tid] += redA[tid + s];
        __syncthreads();
    }
    if (tid == 0 && labels[row] != IGNORE_INDEX) {
        atomicAdd(&accum[0], redA[0]);
    }
}

__global__ void init_accum(float* accum)
{
    if (threadIdx.x < 2) accum[threadIdx.x] = 0.f;
}

__global__ __launch_bounds__(256) void count_valid(
    const long long* __restrict__ labels, float* __restrict__ accum)
{
    __shared__ int c[256];
    int tid = threadIdx.x;
    int cnt = 0;
    for (int i = tid; i < BT; i += 256) cnt += (labels[i] != IGNORE_INDEX) ? 1 : 0;
    c[tid] = cnt;
    __syncthreads();
    for (int s = 128; s > 0; s >>= 1) {
        if (tid < s) c[tid] += c[tid + s];
        __syncthreads();
    }
    if (tid == 0) accum[1] = (float)c[0];
}

__global__ void finalize_loss(const float* __restrict__ accum, float* __restrict__ out)
{
    out[0] = accum[0] / fmaxf(accum[1], 1.0f);
}

// ---------------------------------------------------------------------------
// Launch. ws layout:
//   [0, 256MB)        student logits  (2048*32000 f32)
//   [256MB, 512MB)    teacher logits  (2048*32000 f32)
//   then              accum[2] f32 {sum, count}
// Requires ws_size >= 2*2048*32000*4 + 8 bytes (~524 MB).
// ---------------------------------------------------------------------------
extern "C" void kernel_launch(void* const* d_in, const int* in_sizes, int n_in,
                              void* d_out, int out_size, void* d_ws, size_t ws_size,
                              hipStream_t stream)
{
    const float*     student_input  = (const float*)d_in[0];      // (BT, H_S)
    const float*     student_weight = (const float*)d_in[1];      // (V, H_S)
    const float*     teacher_input  = (const float*)d_in[2];      // (BT, H_T)
    const float*     teacher_weight = (const float*)d_in[3];      // (V, H_T)
    const long long* shift_labels   = (const long long*)d_in[4];  // (BT,) int64
    float* out = (float*)d_out;

    const size_t logits_elems = (size_t)BT * VCAB;
    float* s_logits = (float*)d_ws;
    float* t_logits = s_logits + logits_elems;
    float* accum    = t_logits + logits_elems;

    init_accum<<<1, 64, 0, stream>>>(accum);

    // x = M tiles (fast) so weight tiles get maximal L2 reuse across M.
    dim3 ggrid(BT / 64, VCAB / 64, 1);   // (32, 500)
    dim3 gblock(128, 1, 1);
    gemm_logits_f32<H_S><<<ggrid, gblock, 0, stream>>>(student_input, student_weight, s_logits);
    gemm_logits_f32<H_T><<<ggrid, gblock, 0, stream>>>(teacher_input, teacher_weight, t_logits);

    count_valid<<<1, 256, 0, stream>>>(shift_labels, accum);
    jsd_rows<<<BT, 256, 0, stream>>>(s_logits, t_logits, shift_labels, accum);
    finalize_loss<<<1, 1, 0, stream>>>(accum, out);
}